// MoE_63831803953658
// MI455X (gfx1250) — compile-verified
//
#include <hip/hip_runtime.h>
#include <hip/hip_bf16.h>
#include <math.h>

typedef __attribute__((ext_vector_type(2))) float v2f;
typedef __attribute__((ext_vector_type(8))) float v8f;

constexpr int E = 16;
constexpr int D = 2048;
constexpr int H = 2048;
constexpr int SPLIT  = 8;          // split-K factor: 2*128*8 = 2048 waves/phase
constexpr int NTILE  = H / 16;     // 128 row tiles of 16 rows per expert

// ---------------------------------------------------------------------------
// Kernel 1: gating. logits = Wg@x + bg ; softmax ; top-2 ; normalized gates.
// One block, 16 waves (one expert per wave, wave32 lane-strided dot).
// ---------------------------------------------------------------------------
__global__ __launch_bounds__(512) void moe_gate_kernel(
    const float* __restrict__ x, const float* __restrict__ Wg,
    const float* __restrict__ bg, int* __restrict__ idx_out,
    float* __restrict__ tkg_out)
{
  __shared__ float logits[E];
  const int wave = threadIdx.x >> 5;   // 0..15 == expert
  const int lane = threadIdx.x & 31;
  const float* wrow = Wg + (size_t)wave * D;
  float s = 0.f;
  for (int d = lane; d < D; d += 32) s = fmaf(wrow[d], x[d], s);
  #pragma unroll
  for (int o = 16; o > 0; o >>= 1) s += __shfl_xor(s, o, 32);
  if (lane == 0) logits[wave] = s + bg[wave];
  __syncthreads();
  if (threadIdx.x == 0) {
    float mx = logits[0];
    #pragma unroll
    for (int i = 1; i < E; ++i) mx = fmaxf(mx, logits[i]);
    float g[E]; float sum = 0.f;
    #pragma unroll
    for (int i = 0; i < E; ++i) { g[i] = expf(logits[i] - mx); sum += g[i]; }
    float v1 = -1e30f, v2 = -1e30f; int i1 = 0, i2 = 0;
    #pragma unroll
    for (int i = 0; i < E; ++i) {
      float gi = g[i] / sum;
      if (gi > v1)      { v2 = v1; i2 = i1; v1 = gi; i1 = i; }
      else if (gi > v2) { v2 = gi; i2 = i; }
    }
    float den = v1 + v2 + 1e-6f;
    idx_out[0] = i1; idx_out[1] = i2;
    tkg_out[0] = v1 / den; tkg_out[1] = v2 / den;
  }
}

// ---------------------------------------------------------------------------
// Kernel 2/4: bandwidth-bound GEMV tile via V_WMMA_F32_16X16X4_F32.
// K and vec_stride are compile-time so the inner loop has a constant trip
// count (K/SPLIT/4 = 64): no runtime division, no remainder loop, single
// 8x-unrolled pipelined body.
// Each 32-lane workgroup computes a 16-row tile over one K-slice and writes
// a 16-float partial. A = fp32 weight tile (16x4); B = the 4-wide vector
// chunk replicated across all 16 N columns -> every accumulator column holds
// the same y-tile. Lane 0 ends with y[0..7] in acc, lane 16 with y[8..15].
//   A layout: lane m (<16): row m, K = k,k+1 ; lane m+16: row m, K = k+2,k+3
//   B layout: lanes 0-15: K = k,k+1 ; lanes 16-31: K = k+2,k+3 (all N equal)
// ---------------------------------------------------------------------------
template <int K, int VSTRIDE>
__global__ __launch_bounds__(32) void moe_gemv16_wmma(
    const float* __restrict__ W,     // [E, H, K] row-major weights
    const float* __restrict__ vecb,  // input vector base (x or h)
    const int* __restrict__ idx,     // selected experts [2]
    float* __restrict__ partial)     // [SPLIT][2][H]
{
  const int lane = threadIdx.x;            // 0..31
  const int b    = blockIdx.x;
  const int s    = b % SPLIT;              // K-slice
  const int t    = (b / SPLIT) % NTILE;    // row tile
  const int e    = b / (SPLIT * NTILE);    // expert slot 0/1
  const int eidx = idx[e];

  const float* Wt = W + (size_t)eidx * (size_t)H * K + (size_t)(t * 16) * K;
  const float* v  = vecb + (size_t)e * VSTRIDE;

  const int m    = lane & 15;
  const int koff = (lane >> 4) << 1;       // 0 for lanes 0-15, 2 for 16-31
  const float* wrow = Wt + (size_t)m * K + koff;
  const float* vp   = v + koff;

  constexpr int KC = K / SPLIT;            // 256
  const int k0 = s * KC;

  v8f acc = {0.f, 0.f, 0.f, 0.f, 0.f, 0.f, 0.f, 0.f};
  #pragma unroll 8
  for (int k = k0; k < k0 + KC; k += 4) {
    v2f a  = *(const v2f*)(wrow + k);      // global_load_b64: weight stream
    v2f bb = *(const v2f*)(vp + k);        // global_load_b64: L2-hot vector
    acc = __builtin_amdgcn_wmma_f32_16x16x4_f32(
        /*neg_a=*/false, a, /*neg_b=*/false, bb,
        /*c_mod=*/(short)0, acc, /*reuse_a=*/false, /*reuse_b=*/false);
  }

  float* outp = partial + ((size_t)s * 2 + e) * H + (size_t)t * 16;
  if (lane == 0) {                         // rows 0..7 of the tile
    *(float4*)(outp)     = make_float4(acc[0], acc[1], acc[2], acc[3]);
    *(float4*)(outp + 4) = make_float4(acc[4], acc[5], acc[6], acc[7]);
  } else if (lane == 16) {                 // rows 8..15 of the tile
    *(float4*)(outp + 8)  = make_float4(acc[0], acc[1], acc[2], acc[3]);
    *(float4*)(outp + 12) = make_float4(acc[4], acc[5], acc[6], acc[7]);
  }
}

// ---------------------------------------------------------------------------
// Kernel 3: deterministic split-K reduction + bias + tanh -> h[2][H]
// ---------------------------------------------------------------------------
__global__ __launch_bounds__(256) void moe_tanh_kernel(
    const float* __restrict__ p1, const float* __restrict__ b1,
    const int* __restrict__ idx, float* __restrict__ h)
{
  int j = blockIdx.x * 256 + threadIdx.x;  // 0 .. 2*H-1
  int e = j / H, col = j - e * H;
  float s = 0.f;
  #pragma unroll
  for (int t = 0; t < SPLIT; ++t) s += p1[((size_t)t * 2 + e) * H + col];
  h[j] = tanhf(s + b1[(size_t)idx[e] * H + col]);
}

// ---------------------------------------------------------------------------
// Kernel 5: reduction + bias + gate-weighted combine -> out[H]
// ---------------------------------------------------------------------------
__global__ __launch_bounds__(256) void moe_combine_kernel(
    const float* __restrict__ p2, const float* __restrict__ b2,
    const int* __restrict__ idx, const float* __restrict__ tkg,
    float* __restrict__ out)
{
  int col = blockIdx.x * 256 + threadIdx.x;  // 0..H-1
  float acc = 0.f;
  #pragma unroll
  for (int e = 0; e < 2; ++e) {
    float s = 0.f;
    #pragma unroll
    for (int t = 0; t < SPLIT; ++t) s += p2[((size_t)t * 2 + e) * H + col];
    acc += tkg[e] * (s + b2[(size_t)idx[e] * H + col]);
  }
  out[col] = acc;
}

// ---------------------------------------------------------------------------
extern "C" void kernel_launch(void* const* d_in, const int* in_sizes, int n_in,
                              void* d_out, int out_size, void* d_ws, size_t ws_size,
                              hipStream_t stream)
{
  const float* x  = (const float*)d_in[0];
  const float* Wg = (const float*)d_in[1];
  const float* bg = (const float*)d_in[2];
  const float* W1 = (const float*)d_in[3];
  const float* b1 = (const float*)d_in[4];
  const float* W2 = (const float*)d_in[5];
  const float* b2 = (const float*)d_in[6];
  // d_in[7] = train (eval path: ignored)
  float* out = (float*)d_out;

  // workspace layout (~280 KB total)
  char*  ws   = (char*)d_ws;
  int*   idx  = (int*)ws;                                       // 2 ints
  float* tkg  = (float*)(ws + 16);                              // 2 floats
  float* p1   = (float*)(ws + 256);                             // SPLIT*2*H
  float* hbuf = (float*)(ws + 256 + (size_t)SPLIT*2*H*4);       // 2*H
  float* p2   = (float*)(ws + 256 + (size_t)SPLIT*2*H*4 + (size_t)2*H*4);

  moe_gate_kernel<<<1, 512, 0, stream>>>(x, Wg, bg, idx, tkg);
  moe_gemv16_wmma<D, 0><<<2 * NTILE * SPLIT, 32, 0, stream>>>(W1, x, idx, p1);
  moe_tanh_kernel<<<(2 * H) / 256, 256, 0, stream>>>(p1, b1, idx, hbuf);
  moe_gemv16_wmma<H, H><<<2 * NTILE * SPLIT, 32, 0, stream>>>(W2, hbuf, idx, p2);
  moe_combine_kernel<<<H / 256, 256, 0, stream>>>(p2, b2, idx, out ? tkg : tkg, out);
}